// AspectMemorySummarizer_24472723652706
// MI455X (gfx1250) — compile-verified
//
#include <hip/hip_runtime.h>
#include <hip/hip_bf16.h>
#include <math.h>

// Problem shape (from reference setup_inputs)
#define B_DIM 1024
#define L_DIM 256
#define E_DIM 300
#define A_DIM 30

#define ASP_ROWS 32        // aspects padded to 2 WMMA tiles of 16
#define ASP_STRIDE 304     // padded LDS row stride (floats)
#define WORD_THRES 0.2f
#define NEG_INF_F (-1e9f)

typedef __attribute__((ext_vector_type(2))) float v2f;
typedef __attribute__((ext_vector_type(8))) float v8f;

__global__ __launch_bounds__(512) void
AspectMemorySummarizer_kernel(const int* __restrict__ inputs,
                              const float* __restrict__ w_emb,
                              const float* __restrict__ a_emb,
                              const float* __restrict__ a_weight,
                              float* __restrict__ enc_out,
                              float* __restrict__ attn_out,
                              float* __restrict__ cs_out)
{
    __shared__ float s_asp[ASP_ROWS * ASP_STRIDE]; // aspect rows, rows 30/31 zero
    __shared__ float s_ianorm[ASP_ROWS];           // 1 / max(||a||, eps)
    __shared__ float s_aw[ASP_ROWS];
    __shared__ float s_cmax[L_DIM];                // per-token max thresholded sim
    __shared__ float s_attn[L_DIM];
    __shared__ int   s_tok[L_DIM];
    __shared__ float s_scal[2];                    // [0]=softmax denom, [1]=centroid_score

    const int b    = blockIdx.x;
    const int tid  = threadIdx.x;
    const int lane = tid & 31;
    const int wave = tid >> 5;

    // ---- stage tokens and aspect matrix into LDS ----
    if (tid < L_DIM) s_tok[tid] = inputs[b * L_DIM + tid];
    for (int idx = tid; idx < ASP_ROWS * E_DIM; idx += 512) {
        int r = idx / E_DIM, c = idx - r * E_DIM;
        s_asp[r * ASP_STRIDE + c] = (r < A_DIM) ? a_emb[r * E_DIM + c] : 0.0f;
    }
    __syncthreads();

    if (tid < ASP_ROWS) {
        float s = 0.0f;
        for (int k = 0; k < E_DIM; ++k) { float v = s_asp[tid * ASP_STRIDE + k]; s += v * v; }
        s_ianorm[tid] = 1.0f / fmaxf(sqrtf(s), 1e-8f);
        s_aw[tid]     = (tid < A_DIM) ? a_weight[tid] : 0.0f;
    }
    __syncthreads();

    // ---- Pass A: one wave per 16-token tile; WMMA f32 16x16x4 over K=300 ----
    {
        const int n = lane & 15;              // token within tile (N index)
        const int h = lane >> 4;              // lane half (K offset +2 in B/A layout)
        const int l = wave * 16 + n;          // token position in sequence
        const long tok = (long)s_tok[l];
        const float* xrow = w_emb + tok * (long)E_DIM;

        v8f acc0 = {};   // aspects 0..15
        v8f acc1 = {};   // aspects 16..31 (30/31 are zero pad)
        float ss = 0.0f; // sum of squares of this token's embedding (half per lane)
        const float* arow0 = s_asp + n * ASP_STRIDE;
        const float* arow1 = s_asp + (n + 16) * ASP_STRIDE;
        for (int k0 = 0; k0 < E_DIM; k0 += 4) {
            const int kk = k0 + 2 * h;
            // B 4x16 fp32: VGPR0 lanes0-15 K=k0 / lanes16-31 K=k0+2; VGPR1 = +1
            v2f bv  = *(const v2f*)(xrow  + kk);
            // A 16x4 fp32: M = lane&15, VGPR0 K=k0/k0+2, VGPR1 = +1
            v2f av0 = *(const v2f*)(arow0 + kk);
            v2f av1 = *(const v2f*)(arow1 + kk);
            ss += bv.x * bv.x + bv.y * bv.y;   // lane half h covers k ≡ {2h,2h+1} mod 4
            acc0 = __builtin_amdgcn_wmma_f32_16x16x4_f32(false, av0, false, bv,
                                                         (short)0, acc0, false, false);
            acc1 = __builtin_amdgcn_wmma_f32_16x16x4_f32(false, av1, false, bv,
                                                         (short)0, acc1, false, false);
        }
        // full token norm: combine the two K-phases across lane halves
        ss += __shfl_xor(ss, 16);
        const float inv_xn = 1.0f / fmaxf(sqrtf(ss), 1e-8f);

        // C/D layout: VGPR i holds M = i + 8*h, N = lane&15
        float cm = 0.0f;
        #pragma unroll
        for (int i = 0; i < 8; ++i) {
            const int m0 = i + 8 * h;
            const int m1 = m0 + 16;
            float sim0 = acc0[i] * s_ianorm[m0] * inv_xn;
            float sim1 = acc1[i] * s_ianorm[m1] * inv_xn;
            float t0 = (sim0 > WORD_THRES) ? sim0 * s_aw[m0] : 0.0f;
            float t1 = (sim1 > WORD_THRES) ? sim1 * s_aw[m1] : 0.0f;
            cm = fmaxf(cm, fmaxf(t0, t1));
        }
        cm = fmaxf(cm, __shfl_xor(cm, 16));   // combine M 0..7(+16..23) with 8..15(+24..31)
        if (h == 0) s_cmax[l] = cm;
    }
    __syncthreads();

    // ---- sequence reductions: length, centroid score, softmax (wave 0) ----
    if (wave == 0) {
        float cnt = 0.0f, csum = 0.0f, smax = NEG_INF_F;
        for (int l2 = lane; l2 < L_DIM; l2 += 32) {
            cnt += (s_tok[l2] != 0) ? 1.0f : 0.0f;
            float c  = s_cmax[l2];
            csum += c;
            float sc = (c > 0.0f) ? c : NEG_INF_F;   // sim==0 masked to -1e9
            smax = fmaxf(smax, sc);
        }
        #pragma unroll
        for (int off = 16; off > 0; off >>= 1) {
            cnt  += __shfl_xor(cnt, off);
            csum += __shfl_xor(csum, off);
            smax  = fmaxf(smax, __shfl_xor(smax, off));
        }
        float denom = 0.0f;
        for (int l2 = lane; l2 < L_DIM; l2 += 32) {
            float c  = s_cmax[l2];
            float sc = (c > 0.0f) ? c : NEG_INF_F;
            float e  = __expf(sc - smax);
            s_attn[l2] = e;
            denom += e;
        }
        #pragma unroll
        for (int off = 16; off > 0; off >>= 1) denom += __shfl_xor(denom, off);
        if (lane == 0) {
            s_scal[0] = denom;
            s_scal[1] = csum / (cnt + 1e-5f);
        }
    }
    __syncthreads();

    const float denom  = s_scal[0];
    const float cscore = s_scal[1];
    if (tid < L_DIM) {
        float a = s_attn[tid] / denom;
        s_attn[tid] = a;
        attn_out[b * L_DIM + tid] = a;
    }
    if (tid == 0) cs_out[b] = cscore;
    __syncthreads();

    // ---- Pass B: z[e] = sum_l a[l] * x[l][e]; coalesced re-gather ----
    if (tid < E_DIM) {
        float z = 0.0f;
        for (int l2 = 0; l2 < L_DIM; ++l2) {
            z += s_attn[l2] * w_emb[(long)s_tok[l2] * E_DIM + tid];
        }
        const float gate = (cscore > 1e-4f) ? 1.0f : 0.0f;
        enc_out[b * E_DIM + tid] = z * gate;
    }
}

extern "C" void kernel_launch(void* const* d_in, const int* in_sizes, int n_in,
                              void* d_out, int out_size, void* d_ws, size_t ws_size,
                              hipStream_t stream) {
    (void)in_sizes; (void)n_in; (void)out_size; (void)d_ws; (void)ws_size;
    const int*   inputs   = (const int*)d_in[0];
    const float* w_emb    = (const float*)d_in[1];
    const float* a_emb    = (const float*)d_in[2];
    const float* a_weight = (const float*)d_in[3];

    float* out      = (float*)d_out;
    float* enc_out  = out;                               // [B, E]
    float* attn_out = out + (size_t)B_DIM * E_DIM;       // [B, L]
    float* cs_out   = attn_out + (size_t)B_DIM * L_DIM;  // [B]

    AspectMemorySummarizer_kernel<<<B_DIM, 512, 0, stream>>>(
        inputs, w_emb, a_emb, a_weight, enc_out, attn_out, cs_out);
}